// CameraAwareMemory_19765439496776
// MI455X (gfx1250) — compile-verified
//
#include <hip/hip_runtime.h>
#include <hip/hip_bf16.h>

// ---------------------------------------------------------------------------
// CameraAwareMemory for MI455X (gfx1250, wave32)
//   C=8, M=2048, D=2048, B=256, BG_KNN=50, TEMP=0.05, MOMENTUM=0.2
// d_out = [loss(1 f32), em_new(16384*2048 f32)]
// ---------------------------------------------------------------------------

#define C_CAMS   8
#define M_PROX   2048
#define D_DIM    2048
#define B_BATCH  256
#define NCOLS    (C_CAMS * M_PROX)   // 16384
#define BG_KNN   50
#define INV_TEMP 20.0f
#define MOM      0.2f

typedef __attribute__((ext_vector_type(2))) float v2f;
typedef __attribute__((ext_vector_type(8))) float v8f;

// ------------------- CDNA5 async global->LDS copy (gated) -------------------
#if defined(__HIP_DEVICE_COMPILE__) && __has_builtin(__builtin_amdgcn_global_load_async_to_lds_b128)
#define HAVE_ASYNC_LDS 1
#else
#define HAVE_ASYNC_LDS 0
#endif

#if HAVE_ASYNC_LDS
typedef __attribute__((vector_size(16))) int v4i_b;             // matches builtin proto V4i
typedef __attribute__((address_space(1))) v4i_b* as1_v4i_ptr;   // global v4i*
typedef __attribute__((address_space(3))) v4i_b* as3_v4i_ptr;   // LDS v4i*
#endif

__device__ __forceinline__ void async_copy16(const float* g, float* l) {
#if HAVE_ASYNC_LDS
  float* gg = const_cast<float*>(g);
  __builtin_amdgcn_global_load_async_to_lds_b128(
      (as1_v4i_ptr)gg, (as3_v4i_ptr)l, 0, 0);
#else
  *(float4*)l = *(const float4*)g;
#endif
}

__device__ __forceinline__ void wait_async_done() {
#if HAVE_ASYNC_LDS
#if __has_builtin(__builtin_amdgcn_s_wait_asynccnt)
  __builtin_amdgcn_s_wait_asynccnt(0);
#else
  asm volatile("s_wait_asynccnt 0x0" ::: "memory");
#endif
#endif
}

// ---------------- fixed-order (deterministic) block reductions --------------
__device__ __forceinline__ float blockSumF256(float v, float* red) {
  int tid = threadIdx.x;
  red[tid] = v; __syncthreads();
  for (int s = 128; s > 0; s >>= 1) {
    if (tid < s) red[tid] += red[tid + s];
    __syncthreads();
  }
  float r = red[0]; __syncthreads();
  return r;
}
__device__ __forceinline__ float blockMaxF256(float v, float* red) {
  int tid = threadIdx.x;
  red[tid] = v; __syncthreads();
  for (int s = 128; s > 0; s >>= 1) {
    if (tid < s) red[tid] = fmaxf(red[tid], red[tid + s]);
    __syncthreads();
  }
  float r = red[0]; __syncthreads();
  return r;
}
__device__ __forceinline__ int blockSumI256(int v, int* red) {
  int tid = threadIdx.x;
  red[tid] = v; __syncthreads();
  for (int s = 128; s > 0; s >>= 1) {
    if (tid < s) red[tid] += red[tid + s];
    __syncthreads();
  }
  int r = red[0]; __syncthreads();
  return r;
}

// order-preserving float->uint key (ascending float == ascending uint)
__device__ __forceinline__ unsigned fkey(float f) {
  unsigned u = __float_as_uint(f);
  return u ^ ((unsigned)((int)u >> 31) | 0x80000000u);
}
__device__ __forceinline__ float funkey(unsigned k) {
  unsigned u = (k & 0x80000000u) ? (k ^ 0x80000000u) : ~k;
  return __uint_as_float(u);
}

// ---------------------------------------------------------------------------
// Kernel 1: sims[256,16384] = X[256,2048] @ E[16384,2048]^T  via
//           V_WMMA_F32_16X16X4_F32 (exact f32 matrix path on CDNA5).
//
// Block = 256 threads (8 waves). Block tile 64(M) x 128(N), KC=32, double-
// buffered LDS filled by GLOBAL_LOAD_ASYNC_TO_LDS_B128 (ASYNCcnt) when
// available. Each wave computes a 32x32 sub-tile: 2 A-frags x 2 B-frags ->
// 4 WMMA per k-step (4 WMMA : 4 b64 fragment loads).
// LDS row stride 36 dwords (144B): bank-conflict-free 16-lane row reads
// (gcd(36,64)=4 -> 16 distinct banks) and 16B-aligned b128 stores.
// ---------------------------------------------------------------------------
#define KC     32
#define LDSTR  36
#define NCHUNK (D_DIM / KC)          // 64

__device__ __forceinline__ void load_tiles(const float* __restrict__ X,
                                           const float* __restrict__ E,
                                           int m0, int n0, int k0,
                                           float* lx, float* le, int tid) {
  const int xrow = tid >> 2, xcol = (tid & 3) * 8;   // X: 64 rows x 32 cols
  const int erow = tid >> 1, ecol = (tid & 1) * 16;  // E: 128 rows x 32 cols
  const float* xg = X + (size_t)(m0 + xrow) * D_DIM + k0 + xcol;
  const float* eg = E + (size_t)(n0 + erow) * D_DIM + k0 + ecol;
  float* xl = lx + xrow * LDSTR + xcol;
  float* el = le + erow * LDSTR + ecol;
  async_copy16(xg, xl);
  async_copy16(xg + 4, xl + 4);
  async_copy16(eg, el);
  async_copy16(eg + 4,  el + 4);
  async_copy16(eg + 8,  el + 8);
  async_copy16(eg + 12, el + 12);
}

__global__ __launch_bounds__(256) void cam_sims_gemm(
    const float* __restrict__ X, const float* __restrict__ E,
    float* __restrict__ S) {
  __shared__ float lx[2][64 * LDSTR];    //  2 x  9216 B
  __shared__ float le[2][128 * LDSTR];   //  2 x 18432 B  (55296 B total)
  const int tid  = threadIdx.x;
  const int wave = tid >> 5;
  const int lane = tid & 31;
  const int m0 = blockIdx.y * 64;        // gridDim.y = 4   (B=256)
  const int n0 = blockIdx.x * 128;       // gridDim.x = 128 (NCOLS=16384)
  const int msub = (wave & 1) * 32;
  const int nsub = (wave >> 1) * 32;
  const int r  = lane & 15;
  const int kb = (lane >> 4) * 2;        // lanes 16..31 take k+2,k+3

  v8f c00 = {}, c01 = {}, c10 = {}, c11 = {};

  load_tiles(X, E, m0, n0, 0, lx[0], le[0], tid);   // prologue -> buffer 0

  for (int c = 0; c < NCHUNK; ++c) {
    wait_async_done();                   // my wave's chunk-c copies landed
    __syncthreads();                     // everyone's chunk-c copies landed
    if (c + 1 < NCHUNK)                  // fill other buffer (last read two
      load_tiles(X, E, m0, n0, (c + 1) * KC,        // barriers ago -> safe)
                 lx[(c + 1) & 1], le[(c + 1) & 1], tid);

    const float* bx = lx[c & 1];
    const float* be = le[c & 1];
#pragma unroll
    for (int kk = 0; kk < KC; kk += 4) {
      v2f a0 = *(const v2f*)&bx[(msub + r)      * LDSTR + kk + kb];
      v2f a1 = *(const v2f*)&bx[(msub + 16 + r) * LDSTR + kk + kb];
      v2f b0 = *(const v2f*)&be[(nsub + r)      * LDSTR + kk + kb];
      v2f b1 = *(const v2f*)&be[(nsub + 16 + r) * LDSTR + kk + kb];
      c00 = __builtin_amdgcn_wmma_f32_16x16x4_f32(false, a0, false, b0,
                                                  (short)0, c00, false, false);
      c01 = __builtin_amdgcn_wmma_f32_16x16x4_f32(false, a0, false, b1,
                                                  (short)0, c01, false, false);
      c10 = __builtin_amdgcn_wmma_f32_16x16x4_f32(false, a1, false, b0,
                                                  (short)0, c10, false, false);
      c11 = __builtin_amdgcn_wmma_f32_16x16x4_f32(false, a1, false, b1,
                                                  (short)0, c11, false, false);
    }
  }

  // C/D layout: VGPR j -> M = j + 8*(lane>=16), N = lane&15
  const int mo = m0 + msub + ((lane >> 4) << 3);
  const int nc = n0 + nsub + (lane & 15);
#pragma unroll
  for (int j = 0; j < 8; j++) {
    S[(size_t)(mo + j) * NCOLS + nc]           = c00[j];
    S[(size_t)(mo + j) * NCOLS + nc + 16]      = c01[j];
    S[(size_t)(mo + 16 + j) * NCOLS + nc]      = c10[j];
    S[(size_t)(mo + 16 + j) * NCOLS + nc + 16] = c11[j];
  }
}

// ---------------------------------------------------------------------------
// Kernel 2: per-sample intra-camera cross entropy over its camera block.
// ce[i] = lse(sims[i, c_i*M : (c_i+1)*M] / TEMP) - sims[i, c_i*M + t_i]/TEMP
// ---------------------------------------------------------------------------
__global__ __launch_bounds__(256) void cam_ce_kernel(
    const float* __restrict__ S, const int* __restrict__ targets,
    const int* __restrict__ cams, float* __restrict__ ce) {
  __shared__ float red[256];
  const int i = blockIdx.x;
  const int tid = threadIdx.x;
  const float* row = S + (size_t)i * NCOLS + (size_t)cams[i] * M_PROX;
  float m = -3.0e38f;
  for (int j = tid; j < M_PROX; j += 256) m = fmaxf(m, row[j]);
  m = blockMaxF256(m, red);
  float s = 0.0f;
  for (int j = tid; j < M_PROX; j += 256) s += expf((row[j] - m) * INV_TEMP);
  s = blockSumF256(s, red);
  if (tid == 0) {
    float lt = row[targets[i]];
    ce[i] = (m * INV_TEMP + logf(s)) - lt * INV_TEMP;
  }
}

// ---------------------------------------------------------------------------
// Kernel 3: per-sample hard-negative loss term.
// Exclude the 8 positive columns (col % M == t_i), select top-50 *values*
// via 32-step bitwise binary search on order-preserving keys (registers),
// then al[i] = lse([pos(8), neg(50)]/TEMP) - mean(pos)/TEMP.
// ---------------------------------------------------------------------------
__global__ __launch_bounds__(256) void cam_al_kernel(
    const float* __restrict__ S, const int* __restrict__ targets,
    float* __restrict__ al) {
  __shared__ float redf[256];
  __shared__ int   redi[256];
  __shared__ float posv[C_CAMS];
  const int i = blockIdx.x;
  const int tid = threadIdx.x;
  const float* row = S + (size_t)i * NCOLS;
  const int t = targets[i];

  if (tid < C_CAMS) posv[tid] = row[t + M_PROX * tid];
  __syncthreads();

  unsigned key[64];
  float mneg = -3.0e38f;
#pragma unroll
  for (int j = 0; j < 64; j++) {
    int col = tid + j * 256;
    float x = row[col];
    bool isPos = ((col & (M_PROX - 1)) == t);
    float xm = isPos ? -3.0e38f : x;
    key[j] = fkey(xm);
    mneg = fmaxf(mneg, xm);
  }
  mneg = blockMaxF256(mneg, redf);

  // largest T with count(key >= T) >= 50  ==> T is the 50th-largest key
  unsigned T = 0u;
  for (int bit = 31; bit >= 0; --bit) {
    unsigned cand = T | (1u << bit);
    int c = 0;
#pragma unroll
    for (int j = 0; j < 64; j++) c += (key[j] >= cand) ? 1 : 0;
    c = blockSumI256(c, redi);
    if (c >= BG_KNN) T = cand;
  }
  int q = 0;
#pragma unroll
  for (int j = 0; j < 64; j++) q += (key[j] > T) ? 1 : 0;
  q = blockSumI256(q, redi);                  // strictly-greater count (<50)

  float mp = posv[0];
#pragma unroll
  for (int c = 1; c < C_CAMS; c++) mp = fmaxf(mp, posv[c]);
  const float mL = fmaxf(mneg, mp) * INV_TEMP;  // max logit among cat set

  float s = 0.0f;
#pragma unroll
  for (int j = 0; j < 64; j++)
    if (key[j] > T) s += expf(funkey(key[j]) * INV_TEMP - mL);
  s = blockSumF256(s, redf);

  if (tid == 0) {
    float v50 = funkey(T);
    float Ssum = s + (float)(BG_KNN - q) * expf(v50 * INV_TEMP - mL);
    float psum = 0.0f;
#pragma unroll
    for (int c = 0; c < C_CAMS; c++) {
      Ssum += expf(posv[c] * INV_TEMP - mL);
      psum += posv[c];
    }
    al[i] = (mL + logf(Ssum)) - (psum * (1.0f / C_CAMS)) * INV_TEMP;
  }
}

// ---------------------------------------------------------------------------
// Kernel 4: stream em (128 MB) into d_out+1. Non-temporal stores (the copy
// target is write-once; only 256 rows are re-read by the EMA kernel).
// ---------------------------------------------------------------------------
__global__ __launch_bounds__(256) void cam_copy_em(
    const float4* __restrict__ src, float* __restrict__ dst, long n4) {
  long idx = (long)blockIdx.x * 256 + threadIdx.x;
  long stride = (long)gridDim.x * 256;
  for (long i = idx; i < n4; i += stride) {
    float4 v = src[i];                   // 16B-aligned b128 load
    __builtin_nontemporal_store(v.x, dst + i * 4 + 0);  // dst is +4B offset
    __builtin_nontemporal_store(v.y, dst + i * 4 + 1);
    __builtin_nontemporal_store(v.z, dst + i * 4 + 2);
    __builtin_nontemporal_store(v.w, dst + i * 4 + 3);
  }
}

// ---------------------------------------------------------------------------
// Kernel 5: sequential EMA scan, parallelized per distinct row.
// Block i owns row r_i only if it is the FIRST sample hitting r_i; it then
// walks all samples j>=i with r_j==r_i in order (exact lax.scan semantics).
// ---------------------------------------------------------------------------
__global__ __launch_bounds__(256) void cam_ema_kernel(
    float* __restrict__ EM, const float* __restrict__ X,
    const int* __restrict__ cams, const int* __restrict__ targets) {
  __shared__ int rows[B_BATCH];
  __shared__ float red[256];
  const int i = blockIdx.x;
  const int tid = threadIdx.x;
  rows[tid] = cams[tid] * M_PROX + targets[tid];
  __syncthreads();
  const int r = rows[i];
  bool first = true;
  for (int j = 0; j < i; j++) first = first && (rows[j] != r);
  if (!first) return;  // uniform exit across block

  float e[8];
#pragma unroll
  for (int k = 0; k < 8; k++) e[k] = EM[(size_t)r * D_DIM + tid + k * 256];

  for (int j = i; j < B_BATCH; j++) {
    if (rows[j] != r) continue;          // uniform (shared rows[])
    float ss = 0.0f;
#pragma unroll
    for (int k = 0; k < 8; k++) {
      e[k] = MOM * e[k] + (1.0f - MOM) * X[(size_t)j * D_DIM + tid + k * 256];
      ss += e[k] * e[k];
    }
    ss = blockSumF256(ss, red);          // broadcast to all threads
    float inv = 1.0f / sqrtf(ss);
#pragma unroll
    for (int k = 0; k < 8; k++) e[k] *= inv;
  }
#pragma unroll
  for (int k = 0; k < 8; k++) EM[(size_t)r * D_DIM + tid + k * 256] = e[k];
}

// ---------------------------------------------------------------------------
// Kernel 6: deterministic final loss reduction.
// loss = sum_i ce[i]/cnt[c_i] + (epoch>=5) * 0.5*al[i]/cnt[c_i]
// ---------------------------------------------------------------------------
__global__ __launch_bounds__(256) void cam_loss_kernel(
    const float* __restrict__ ce, const float* __restrict__ al,
    const int* __restrict__ cams, const int* __restrict__ epoch,
    float* __restrict__ out_loss) {
  __shared__ float red[256];
  __shared__ int cnt[C_CAMS];
  const int tid = threadIdx.x;
  if (tid < C_CAMS) {
    int c = 0;
    for (int j = 0; j < B_BATCH; j++) c += (cams[j] == tid) ? 1 : 0;
    cnt[tid] = (c > 0) ? c : 1;
  }
  __syncthreads();
  const int ep = epoch[0];
  float inv = 1.0f / (float)cnt[cams[tid]];
  float term = ce[tid] * inv;
  if (ep >= 5) term += 0.5f * al[tid] * inv;
  float s = blockSumF256(term, red);
  if (tid == 0) out_loss[0] = s;
}

// ---------------------------------------------------------------------------
extern "C" void kernel_launch(void* const* d_in, const int* in_sizes, int n_in,
                              void* d_out, int out_size, void* d_ws, size_t ws_size,
                              hipStream_t stream) {
  const float* X       = (const float*)d_in[0];  // [256, 2048]
  const float* EM      = (const float*)d_in[1];  // [16384, 2048]
  const int*   targets = (const int*)d_in[2];    // [256]
  const int*   cams    = (const int*)d_in[3];    // [256]
  const int*   epoch   = (const int*)d_in[4];    // [1]

  float* out      = (float*)d_out;
  float* out_loss = out;                         // [1]
  float* out_em   = out + 1;                     // [16384*2048]

  float* sims = (float*)d_ws;                    // 256*16384 f32 = 16 MB
  float* ce   = sims + (size_t)B_BATCH * NCOLS;  // 256 f32
  float* al   = ce + B_BATCH;                    // 256 f32

  cam_sims_gemm<<<dim3(NCOLS / 128, B_BATCH / 64), 256, 0, stream>>>(X, EM, sims);
  cam_ce_kernel<<<B_BATCH, 256, 0, stream>>>(sims, targets, cams, ce);
  cam_al_kernel<<<B_BATCH, 256, 0, stream>>>(sims, targets, al);
  cam_copy_em<<<8192, 256, 0, stream>>>((const float4*)EM, out_em,
                                        (long)NCOLS * D_DIM / 4);
  cam_ema_kernel<<<B_BATCH, 256, 0, stream>>>(out_em, X, cams, targets);
  cam_loss_kernel<<<1, 256, 0, stream>>>(ce, al, cams, epoch, out_loss);
}